// TripletHardMarginLoss_81767587381280
// MI455X (gfx1250) — compile-verified
//
#include <hip/hip_runtime.h>
#include <math.h>

// Problem constants from the reference.
#define B_SZ 8192
#define D_SZ 128
#define C_SZ 64

typedef __attribute__((ext_vector_type(2))) float v2f;
typedef __attribute__((ext_vector_type(8))) float v8f;

// ---------------------------------------------------------------------------
// Kernel 1: per-row squared norms sq[i] = sum_k e[i][k]^2
// ---------------------------------------------------------------------------
__global__ __launch_bounds__(256) void sq_kernel(const float* __restrict__ e,
                                                 float* __restrict__ sq) {
    int i = blockIdx.x * blockDim.x + threadIdx.x;
    if (i >= B_SZ) return;
    const float4* r = (const float4*)(e + (size_t)i * D_SZ);
    float s = 0.0f;
#pragma unroll
    for (int k = 0; k < D_SZ / 4; ++k) {
        float4 v = r[k];
        s += v.x * v.x + v.y * v.y + v.z * v.z + v.w * v.w;
    }
    sq[i] = s;
}

// ---------------------------------------------------------------------------
// Kernel 2: fused fp32-WMMA Gram matrix + batch-hard mining.
// Each wave32 owns a 16-row strip and sweeps all 8192 columns, TWO 16-wide
// column tiles per iteration (independent WMMA accumulation chains) using
// V_WMMA_F32_16X16X4_F32 (K stepped by 4 -> 2x32 WMMAs per iteration).
//
// fp32 WMMA operand layouts (ISA 7.12.2):
//   A (16x4, MxK): lane&15 = M; VGPR0 = K {0 | 2}, VGPR1 = K {1 | 3} for
//                  lanes 0-15 | 16-31.  -> lane loads A[m][k0 + 2*hi + {0,1}]
//   B (4x16, KxN): mirrored -> lane loads B[k0 + 2*hi + {0,1}][n=lane&15],
//                  and since B = e^T, that is e[col n][k0 + 2*hi + {0,1}].
//   C/D (16x16):   VGPR v holds row m = v + 8*hi, column n = lane&15.
// ---------------------------------------------------------------------------
__global__ __launch_bounds__(128) void mine_kernel(const float* __restrict__ e,
                                                   const int*   __restrict__ labels,
                                                   const float* __restrict__ sq,
                                                   int* __restrict__ hp,
                                                   int* __restrict__ hn) {
    const int lane    = threadIdx.x & 31;
    const int wave    = threadIdx.x >> 5;
    const int l15     = lane & 15;
    const int hi      = lane >> 4;                      // 0 or 1 (half-wave)
    const int rowBase = (blockIdx.x * 4 + wave) * 16;   // 16 rows per wave

    // Preload the A fragments for this 16-row strip (K = 128 -> 32 k-steps).
    v2f a[32];
    const float* arow = e + (size_t)(rowBase + l15) * D_SZ + 2 * hi;
#pragma unroll
    for (int kt = 0; kt < 32; ++kt) {
        a[kt].x = arow[kt * 4 + 0];
        a[kt].y = arow[kt * 4 + 1];
    }

    // Per-(lane,slot) row metadata: slot v covers global row rowBase + v + 8*hi.
    float rsq[8];
    int   rlab[8];
#pragma unroll
    for (int v = 0; v < 8; ++v) {
        int gm  = rowBase + v + 8 * hi;
        rsq[v]  = sq[gm];
        rlab[v] = labels[gm];
    }

    // Running masked argmax (hardest positive) / argmin (hardest negative).
    float pmax[8], nmin[8];
    int   pidx[8], nidx[8];
#pragma unroll
    for (int v = 0; v < 8; ++v) {
        pmax[v] = -1e30f; pidx[v] = -1;
        nmin[v] =  1e30f; nidx[v] = -1;
    }

    for (int colBase = 0; colBase < B_SZ; colBase += 32) {
        const int gn0 = colBase + l15;
        const int gn1 = colBase + 16 + l15;
        const float* brow0 = e + (size_t)gn0 * D_SZ + 2 * hi;
        const float* brow1 = e + (size_t)gn1 * D_SZ + 2 * hi;

        v8f acc0 = {};
        v8f acc1 = {};
#pragma unroll
        for (int kt = 0; kt < 32; ++kt) {
            v2f b0, b1;
            b0.x = brow0[kt * 4 + 0];
            b0.y = brow0[kt * 4 + 1];
            b1.x = brow1[kt * 4 + 0];
            b1.y = brow1[kt * 4 + 1];
            // Two independent accumulation chains sharing the A fragment.
            acc0 = __builtin_amdgcn_wmma_f32_16x16x4_f32(
                false, a[kt], false, b0, (short)0, acc0, false, false);
            acc1 = __builtin_amdgcn_wmma_f32_16x16x4_f32(
                false, a[kt], false, b1, (short)0, acc1, false, false);
        }

        const int   clab0 = labels[gn0];
        const float csq0  = sq[gn0];
        const int   clab1 = labels[gn1];
        const float csq1  = sq[gn1];

#pragma unroll
        for (int v = 0; v < 8; ++v) {
            const int gm = rowBase + v + 8 * hi;

            float d2 = rsq[v] + csq0 - 2.0f * acc0[v];
            d2 = fmaxf(d2, 0.0f);
            if (rlab[v] == clab0) {
                // positive candidate (exclude diagonal); strict > keeps the
                // first (lowest-index) maximum within this lane's columns
                if (gn0 != gm && d2 > pmax[v]) { pmax[v] = d2; pidx[v] = gn0; }
            } else {
                if (d2 < nmin[v]) { nmin[v] = d2; nidx[v] = gn0; }
            }

            float e2 = rsq[v] + csq1 - 2.0f * acc1[v];
            e2 = fmaxf(e2, 0.0f);
            if (rlab[v] == clab1) {
                if (gn1 != gm && e2 > pmax[v]) { pmax[v] = e2; pidx[v] = gn1; }
            } else {
                if (e2 < nmin[v]) { nmin[v] = e2; nidx[v] = gn1; }
            }
        }
    }

    // Reduce each row across its 16-lane group (xor masks stay inside group).
#pragma unroll
    for (int v = 0; v < 8; ++v) {
        float pm = pmax[v]; int pi = pidx[v];
        float nm = nmin[v]; int ni = nidx[v];
#pragma unroll
        for (int m = 1; m < 16; m <<= 1) {
            float opm = __shfl_xor(pm, m, 32); int opi = __shfl_xor(pi, m, 32);
            float onm = __shfl_xor(nm, m, 32); int oni = __shfl_xor(ni, m, 32);
            // max with tie -> smaller column index (JAX argmax = first hit)
            if (opi >= 0 && (opm > pm || (opm == pm && (pi < 0 || opi < pi)))) {
                pm = opm; pi = opi;
            }
            // min with tie -> smaller column index
            if (oni >= 0 && (onm < nm || (onm == nm && (ni < 0 || oni < ni)))) {
                nm = onm; ni = oni;
            }
        }
        if (l15 == 0) {
            int gm = rowBase + v + 8 * hi;
            hp[gm] = pi;   // -1 => no positive
            hn[gm] = ni;   // -1 => no negative
        }
    }
}

// ---------------------------------------------------------------------------
// Kernel 3: exact norms on the mined indices + final masked-mean loss.
// Single block; ~2 MFLOP tail.
// ---------------------------------------------------------------------------
__global__ __launch_bounds__(256) void loss_kernel(const float* __restrict__ e,
                                                   const float* __restrict__ mm,
                                                   const int*   __restrict__ labels,
                                                   const int*   __restrict__ hp,
                                                   const int*   __restrict__ hn,
                                                   float* __restrict__ out) {
    __shared__ float ssum[256];
    __shared__ float scnt[256];
    float lsum = 0.0f, lcnt = 0.0f;

    for (int i = threadIdx.x; i < B_SZ; i += 256) {
        int p = hp[i], n = hn[i];
        if (p >= 0 && n >= 0) {
            const float4* ri = (const float4*)(e + (size_t)i * D_SZ);
            const float4* rp = (const float4*)(e + (size_t)p * D_SZ);
            const float4* rn = (const float4*)(e + (size_t)n * D_SZ);
            float dap2 = 0.0f, dan2 = 0.0f;
#pragma unroll 8
            for (int k = 0; k < D_SZ / 4; ++k) {
                float4 av = ri[k], pv = rp[k], nv = rn[k];
                float dx = av.x - pv.x, dy = av.y - pv.y,
                      dz = av.z - pv.z, dw = av.w - pv.w;
                dap2 += dx * dx + dy * dy + dz * dz + dw * dw;
                dx = av.x - nv.x; dy = av.y - nv.y;
                dz = av.z - nv.z; dw = av.w - nv.w;
                dan2 += dx * dx + dy * dy + dz * dz + dw * dw;
            }
            float margin = mm[labels[i] * C_SZ + labels[n]];
            float loss = fmaxf(sqrtf(dap2) - sqrtf(dan2) + margin, 0.0f);
            lsum += loss;
            lcnt += 1.0f;
        }
    }

    ssum[threadIdx.x] = lsum;
    scnt[threadIdx.x] = lcnt;
    __syncthreads();
    for (int s = 128; s > 0; s >>= 1) {
        if ((int)threadIdx.x < s) {
            ssum[threadIdx.x] += ssum[threadIdx.x + s];
            scnt[threadIdx.x] += scnt[threadIdx.x + s];
        }
        __syncthreads();
    }
    if (threadIdx.x == 0) out[0] = ssum[0] / fmaxf(scnt[0], 1.0f);
}

// ---------------------------------------------------------------------------
// Host launcher. Inputs: [0] embeddings f32[B,D], [1] margin_matrix f32[C,C],
// [2] labels int[B]. Output: scalar f32. Workspace: hp[B], hn[B] (int),
// sq[B] (float) = 96 KB.
// ---------------------------------------------------------------------------
extern "C" void kernel_launch(void* const* d_in, const int* in_sizes, int n_in,
                              void* d_out, int out_size, void* d_ws, size_t ws_size,
                              hipStream_t stream) {
    const float* emb    = (const float*)d_in[0];
    const float* mm     = (const float*)d_in[1];
    const int*   labels = (const int*)d_in[2];
    float*       out    = (float*)d_out;

    int*   hp = (int*)d_ws;
    int*   hn = hp + B_SZ;
    float* sq = (float*)(hn + B_SZ);

    sq_kernel<<<B_SZ / 256, 256, 0, stream>>>(emb, sq);
    // 128 threads = 4 waves per block, 16 rows per wave -> 64 rows per block.
    mine_kernel<<<B_SZ / 64, 128, 0, stream>>>(emb, labels, sq, hp, hn);
    loss_kernel<<<1, 256, 0, stream>>>(emb, mm, labels, hp, hn, out);
}